// LengthRegulator_41085657153547
// MI455X (gfx1250) — compile-verified
//
#include <hip/hip_runtime.h>

// LengthRegulator forward for MI455X (gfx1250).
// out[b,j,:] = x[b, lower_bound(cum[b,:], j), :]  (zero past lengths[b])
// lengths[b] = cum[b, TX-1]
//
// Memory-bound gather (~29 MB traffic -> ~1.5-3 us at 23.3 TB/s). The
// reference einsum multiplies by a one-hot matrix; a WMMA formulation would
// add ~7.5 GFLOP of multiply-by-0/1 and still move the same bytes. Optimal
// CDNA5 path = data movement: async global->LDS DMA for the cum row,
// b128 loads of x (L2-resident, reused), non-temporal b128 stores of out
// (write-once stream, keep it out of the caches).

#define B_  16
#define TX_ 512
#define D_  256

typedef float v4f __attribute__((ext_vector_type(4)));

// ---------------------------------------------------------------------------
// Kernel 1: per-batch inclusive cumsum of repeat_count.
// Two-level wave32 scan: shfl_up intra-wave, wave 0 scans the 16 wave sums.
// Only 2 barriers (vs 27 for Hillis-Steele in LDS).
// Also writes lengths[b] (as float) into the tail of d_out.
// ---------------------------------------------------------------------------
__global__ __launch_bounds__(TX_) void lr_cumsum_kernel(const int* __restrict__ rc,
                                                        int* __restrict__ cum,
                                                        float* __restrict__ out,
                                                        int Ty) {
    __shared__ int wsum[TX_ / 32];
    const int b    = blockIdx.x;
    const int t    = threadIdx.x;
    const int wave = t >> 5;
    const int lane = t & 31;

    int v = rc[b * TX_ + t];

    // inclusive scan within wave32
#pragma unroll
    for (int off = 1; off < 32; off <<= 1) {
        int n = __shfl_up(v, off, 32);
        if (lane >= off) v += n;
    }
    if (lane == 31) wsum[wave] = v;
    __syncthreads();

    // wave 0 scans the 16 wave totals
    if (wave == 0) {
        int w = (lane < TX_ / 32) ? wsum[lane] : 0;
#pragma unroll
        for (int off = 1; off < TX_ / 32; off <<= 1) {
            int n = __shfl_up(w, off, 32);
            if (lane >= off) w += n;
        }
        if (lane < TX_ / 32) wsum[lane] = w;
    }
    __syncthreads();

    if (wave > 0) v += wsum[wave - 1];

    cum[b * TX_ + t] = v;
    if (t == TX_ - 1) {
        // outputs are concatenated flat: expanded (B*Ty*D) then lengths (B)
        out[(size_t)B_ * (size_t)Ty * D_ + b] = (float)v;
    }
}

// ---------------------------------------------------------------------------
// Kernel 2: gather. 8 wave32s per block, one output frame per wave.
// cum row staged into LDS via gfx1250 async DMA (ASYNCcnt-tracked).
// ---------------------------------------------------------------------------
__global__ __launch_bounds__(256) void lr_gather_kernel(const float* __restrict__ x,
                                                        const int* __restrict__ cum,
                                                        float* __restrict__ out,
                                                        int Ty) {
    __shared__ __attribute__((aligned(16))) int s_cum[TX_];
    const int b   = blockIdx.y;
    const int tid = threadIdx.x;

    // Stage this batch's inclusive cumsum row (2 KB) into LDS with the
    // CDNA5 async global->LDS DMA path (no VGPR round trip).
    if (tid < TX_ / 4) {
        const int* gsrc = cum + b * TX_ + tid * 4;
        unsigned lds_off = (unsigned)(size_t)&s_cum[tid * 4];  // low 32b of flat LDS addr == LDS offset
        asm volatile("global_load_async_to_lds_b128 %0, %1, off"
                     :: "v"(lds_off), "v"(gsrc)
                     : "memory");
    }
    asm volatile("s_wait_asynccnt 0" ::: "memory");
    __syncthreads();

    const int wave = tid >> 5;   // wave32: 8 waves per 256-thread block
    const int lane = tid & 31;
    const int j = blockIdx.x * 8 + wave;
    if (j >= Ty) return;

    v4f* dst = reinterpret_cast<v4f*>(out + ((size_t)b * Ty + j) * D_);

    const int total = s_cum[TX_ - 1];
    if (j >= total) {
        // past this batch's length: reference result is zero
        v4f z = {0.f, 0.f, 0.f, 0.f};
        __builtin_nontemporal_store(z, dst + lane);
        __builtin_nontemporal_store(z, dst + lane + 32);
        return;
    }

    // lower_bound: first i with cum[i] > j (phoneme owning frame j).
    // All 32 lanes search the same path -> LDS bank broadcast, conflict-free.
    int lo = 0, hi = TX_;
    while (lo < hi) {
        int mid = (lo + hi) >> 1;
        if (s_cum[mid] > j) hi = mid; else lo = mid + 1;
    }

    // Copy one 1 KB row: 64 float4s, 32 lanes x 2 b128 transactions each.
    // x loads stay regular-temporal (row reused ~MAX_DUR/2 times, L2-resident);
    // out stores are non-temporal (write-once stream).
    const v4f* src = reinterpret_cast<const v4f*>(x + ((size_t)b * TX_ + lo) * D_);
    v4f a0 = src[lane];
    v4f a1 = src[lane + 32];
    __builtin_nontemporal_store(a0, dst + lane);
    __builtin_nontemporal_store(a1, dst + lane + 32);
}

// ---------------------------------------------------------------------------
extern "C" void kernel_launch(void* const* d_in, const int* in_sizes, int n_in,
                              void* d_out, int out_size, void* d_ws, size_t ws_size,
                              hipStream_t stream) {
    const float* x  = (const float*)d_in[0];   // (B, TX, D) float32
    const int*   rc = (const int*)d_in[1];     // (B, TX) int32
    float* out = (float*)d_out;                // B*Ty*D floats + B lengths
    int*   cum = (int*)d_ws;                   // (B, TX) workspace

    // out_size = B*Ty*D + B  =>  recover data-dependent Ty
    int Ty = (out_size - B_) / (B_ * D_);
    if (Ty <= 0) return;

    lr_cumsum_kernel<<<dim3(B_), dim3(TX_), 0, stream>>>(rc, cum, out, Ty);

    dim3 grid((Ty + 7) / 8, B_);
    lr_gather_kernel<<<grid, dim3(256), 0, stream>>>(x, cum, out, Ty);
}